// Net_20658792694324
// MI455X (gfx1250) — compile-verified
//
#include <hip/hip_runtime.h>
#include <math.h>
#include <stdint.h>

// ---------------- problem constants (match reference) ----------------
#define BB   256   // batch (graphs)
#define NN   512   // nodes layer 1
#define FF   128   // input features
#define HH   128   // hidden
#define CC   10    // classes
#define K1   256
#define K2   128
#define K3   64

typedef __attribute__((ext_vector_type(2))) float v2f;
typedef __attribute__((ext_vector_type(8))) float v8f;
typedef __attribute__((ext_vector_type(4))) unsigned int v4u;
typedef __attribute__((ext_vector_type(8))) int v8i;
typedef __attribute__((ext_vector_type(4))) int v4i;

#if defined(__has_builtin)
#  if __has_builtin(__builtin_amdgcn_tensor_load_to_lds) && \
      __has_builtin(__builtin_amdgcn_s_wait_tensorcnt)
#    define USE_TDM 1
#  endif
#endif
#ifndef USE_TDM
#  define USE_TDM 0
#endif

__device__ __forceinline__ float wave_reduce_sum(float v) {
  #pragma unroll
  for (int off = 16; off > 0; off >>= 1) v += __shfl_xor(v, off, 32);
  return v;
}

#if USE_TDM
// ---- TDM: load a (rows x 16) f32 tile, row stride = stride_elems, into LDS.
// D# layout per CDNA5 ISA (group0) / §8.4 (group1).
// 6-arg builtin form (amdgpu-toolchain clang): (v4u, v8i, v4i, v4i, v8i, i32).
__device__ __forceinline__ void tdm_load_panel(unsigned lds_off, const float* gptr,
                                               int rows, int stride_elems) {
  const unsigned long long ga = (unsigned long long)(uintptr_t)gptr;
  v4u g0;
  g0[0] = 1u;                                         // count=1, user descriptor
  g0[1] = lds_off;                                    // lds_addr (bytes)
  g0[2] = (unsigned)(ga & 0xFFFFFFFFull);             // global_addr[31:0]
  g0[3] = (unsigned)((ga >> 32) & 0x1FFFFFFull)       // global_addr[56:32]
        | (2u << 30);                                 // type = 2 ("image")
  const unsigned td0 = (unsigned)stride_elems;        // tensor_dim0 (>= tile width)
  const unsigned td1 = (unsigned)rows;                // tensor_dim1
  v8i g1;
  g1[0] = (int)(2u << 16);                            // mask=0, data_size=2 (4B)
  g1[1] = (int)(td0 << 16);                           // barrier_addr=0 | td0[15:0]
  g1[2] = (int)((td0 >> 16) | (td1 << 16));           // td0[31:16] | td1[15:0]
  g1[3] = (int)((td1 >> 16) | (16u << 16));           // td1[31:16] | tile_dim0=16
  g1[4] = (int)(td1 & 0xFFFFu);                       // tile_dim1=rows, tile_dim2=0
  g1[5] = (int)(unsigned)stride_elems;                // tensor_dim0_stride[31:0]
  g1[6] = 0;                                          // stride0[47:32] | stride1 lo
  g1[7] = 0;                                          // stride1 hi
  v4i z4 = {0, 0, 0, 0};                              // groups 2/3 unused (2D tile)
  v8i z8 = {0, 0, 0, 0, 0, 0, 0, 0};
  __builtin_amdgcn_tensor_load_to_lds(g0, g1, z4, z4, z8, 0);
}
#endif

// ---------------- WMMA f32 GEMM ---------------------------------------
// Block = 128 threads (4 waves). Block tile = 64(M) x 16(N); wave w owns the
// 16x16 tile at rows m0+16w. The K x 16 B panel is staged once in LDS (TDM
// async-tensor load when available), then each wave streams its A rows from
// global (adjacency / features) and issues V_WMMA_F32_16X16X4_F32.
// EPI 0: C = dinv[row] * acc                       (T = dinv (.) XW)
// EPI 1: C = dinv[row]*(acc + tself[row,col]) + bias[col], optional leaky
enum { EPI_SCALE_ROW = 0, EPI_GCN = 1 };

template<bool A_IS_INT, int EPI, bool LEAKY>
__global__ __launch_bounds__(128)
void wmma_gemm_kernel(const void* __restrict__ Aptr,
                      const float* __restrict__ Bptr,
                      float* __restrict__ Cptr,
                      int M, int K, int N,
                      long strideA, long strideB, long strideC,
                      const float* __restrict__ dinv,
                      const float* __restrict__ tself,
                      const float* __restrict__ bias)
{
  __shared__ float bpanel[NN * 16];     // up to 512 x 16 f32 = 32 KiB

  const int lane = threadIdx.x & 31;
  const int wv   = threadIdx.x >> 5;    // 0..3
  const int hi   = lane >> 4;           // 0 | 1
  const int lo   = lane & 15;
  const int n0   = blockIdx.x * 16;
  const int m0   = blockIdx.y * 64 + wv * 16;
  const int b    = blockIdx.z;

  const float* Bb = Bptr + (strideB ? (long)b * strideB : 0);

  // ---- stage B panel (K x 16 at column n0) into LDS ----
#if USE_TDM
  if ((threadIdx.x >> 5) == 0) {        // one wave issues the DMA (EXEC ignored)
    tdm_load_panel((unsigned)(uintptr_t)bpanel, Bb + n0, K, N);
    __builtin_amdgcn_s_wait_tensorcnt(0);
  }
  __syncthreads();
#else
  for (int i = threadIdx.x; i < K * 16; i += 128)
    bpanel[i] = Bb[(long)(i >> 4) * N + n0 + (i & 15)];
  __syncthreads();
#endif

  const int row = m0 + lo;              // A row owned by this lane
  v8f acc = {};

  if (A_IS_INT) {
    const int* Ar = (const int*)Aptr + (long)b * strideA + (long)row * K;
    for (int k0 = 0; k0 < K; k0 += 4) {
      const int k = k0 + (hi << 1);
      v2f av; av.x = (float)Ar[k]; av.y = (float)Ar[k + 1];
      v2f bv; bv.x = bpanel[k * 16 + lo];
              bv.y = bpanel[(k + 1) * 16 + lo];
      acc = __builtin_amdgcn_wmma_f32_16x16x4_f32(false, av, false, bv,
                                                  (short)0, acc, false, false);
    }
  } else {
    const float* Ar = (const float*)Aptr + (long)b * strideA + (long)row * K;
    for (int k0 = 0; k0 < K; k0 += 4) {
      const int k = k0 + (hi << 1);
      v2f av; av.x = Ar[k]; av.y = Ar[k + 1];
      v2f bv; bv.x = bpanel[k * 16 + lo];
              bv.y = bpanel[(k + 1) * 16 + lo];
      acc = __builtin_amdgcn_wmma_f32_16x16x4_f32(false, av, false, bv,
                                                  (short)0, acc, false, false);
    }
  }

  float* Cb = Cptr + (long)b * strideC;
  #pragma unroll
  for (int v = 0; v < 8; ++v) {
    const int r = m0 + v + (hi << 3);
    const int c = n0 + lo;
    float val = acc[v];
    if (EPI == EPI_SCALE_ROW) {
      val *= dinv[(long)b * M + r];
    } else {
      const float dv = dinv[(long)b * M + r];
      val = dv * (val + tself[((long)b * M + r) * (long)N + c]) + bias[c];
      if (LEAKY) val = val > 0.f ? val : 0.01f * val;
    }
    Cb[(long)r * N + c] = val;
  }
}

// ---------------- degree -> dinv = rsqrt(1 + rowsum(A)) --------------
template<bool A_IS_INT>
__global__ void degree_kernel(const void* __restrict__ A, float* __restrict__ dinv, int n) {
  const int wave = (int)((blockIdx.x * blockDim.x + threadIdx.x) >> 5); // = b*n + i
  const int lane = threadIdx.x & 31;
  float sum = 0.f;
  if (A_IS_INT) {
    const int* Ar = (const int*)A + (long)wave * n;
    int s = 0;
    for (int j = lane; j < n; j += 32) s += Ar[j];
    sum = (float)s;
  } else {
    const float* Ar = (const float*)A + (long)wave * n;
    for (int j = lane; j < n; j += 32) sum += Ar[j];
  }
  sum = wave_reduce_sum(sum);
  if (lane == 0) dinv[wave] = rsqrtf(sum + 1.0f);
}

// ---------------- sv[row] = dot(h[row, 0:128], w[0:128]) -------------
__global__ void matvec_kernel(const float* __restrict__ X, const float* __restrict__ w,
                              float* __restrict__ out) {
  const int wave = (int)((blockIdx.x * blockDim.x + threadIdx.x) >> 5);
  const int lane = threadIdx.x & 31;
  const float* xr = X + (long)wave * HH;
  float s = xr[lane] * w[lane] + xr[lane + 32] * w[lane + 32]
          + xr[lane + 64] * w[lane + 64] + xr[lane + 96] * w[lane + 96];
  s = wave_reduce_sum(s);
  if (lane == 0) out[wave] = s;
}

// ---- s[i] = dinv[i]*( sum_j A[i,j]*dinv[j]*sv[j] + dinv[i]*sv[i] ) + bp
template<bool A_IS_INT>
__global__ void score_prop_kernel(const void* __restrict__ A, const float* __restrict__ dinv,
                                  const float* __restrict__ sv, const float* __restrict__ bp,
                                  float* __restrict__ s, int n) {
  const int wave = (int)((blockIdx.x * blockDim.x + threadIdx.x) >> 5); // b*n + i
  const int lane = threadIdx.x & 31;
  const int b = wave / n, i = wave - b * n;
  const float* dv  = dinv + (long)b * n;
  const float* svb = sv   + (long)b * n;
  float sum = 0.f;
  if (A_IS_INT) {
    const int* Ar = (const int*)A + (long)wave * n;
    for (int j = lane; j < n; j += 32) if (Ar[j]) sum += dv[j] * svb[j];
  } else {
    const float* Ar = (const float*)A + (long)wave * n;
    for (int j = lane; j < n; j += 32) sum += Ar[j] * dv[j] * svb[j];
  }
  sum = wave_reduce_sum(sum);
  if (lane == 0) s[wave] = dv[i] * (sum + dv[i] * svb[i]) + bp[0];
}

// ---------------- exact top-k by rank counting (O(n^2) per graph) ----
__global__ void topk_kernel(const float* __restrict__ s, int n, int k,
                            int* __restrict__ idx, float* __restrict__ val) {
  __shared__ float ss[NN];
  const int b = blockIdx.x;
  const float* sb = s + (long)b * n;
  for (int i = threadIdx.x; i < n; i += blockDim.x) ss[i] = sb[i];
  __syncthreads();
  for (int i = threadIdx.x; i < n; i += blockDim.x) {
    const float my = ss[i];
    int rank = 0;
    for (int j = 0; j < n; ++j) {
      const float oj = ss[j];
      rank += (oj > my) || (oj == my && j < i);   // stable descending rank
    }
    if (rank < k) { idx[(long)b * k + rank] = i; val[(long)b * k + rank] = my; }
  }
}

// ---------------- xk[b,p,:] = h[b, idx[p], :] * tanh(val[p]) ---------
__global__ void gather_x_kernel(const float* __restrict__ hsrc, const int* __restrict__ idx,
                                const float* __restrict__ val, float* __restrict__ xk,
                                int n, int k) {
  const int p = blockIdx.x % k, b = blockIdx.x / k;
  const int h = threadIdx.x;                       // 128 threads
  const int id = idx[(long)b * k + p];
  const float t = tanhf(val[(long)b * k + p]);
  xk[((long)b * k + p) * HH + h] = hsrc[((long)b * n + id) * HH + h] * t;
}

// ---------------- ak[b,p,q] = A[b, idx[p], idx[q]] -------------------
template<bool A_IS_INT>
__global__ void gather_a_kernel(const void* __restrict__ Asrc, const int* __restrict__ idx,
                                float* __restrict__ Adst, int n, int k) {
  const int b = blockIdx.y, p = blockIdx.x;
  const int ip = idx[(long)b * k + p];
  for (int q = threadIdx.x; q < k; q += blockDim.x) {
    const int iq = idx[(long)b * k + q];
    float v;
    if (A_IS_INT) v = (float)((const int*)Asrc)[((long)b * n + ip) * n + iq];
    else          v = ((const float*)Asrc)[((long)b * n + ip) * n + iq];
    Adst[((long)b * k + p) * k + q] = v;
  }
}

// ---------------- o[b,:] += [mean_p xk, max_p xk] --------------------
__global__ void readout_kernel(const float* __restrict__ xk, float* __restrict__ o, int k) {
  const int b = blockIdx.x, h = threadIdx.x;       // 128 threads
  const float* xb = xk + (long)b * k * HH;
  float sum = 0.f, mx = -3.4e38f;
  for (int p = 0; p < k; ++p) { float v = xb[(long)p * HH + h]; sum += v; mx = fmaxf(mx, v); }
  o[(long)b * (2 * HH) + h]      += sum / (float)k;
  o[(long)b * (2 * HH) + HH + h] += mx;
}

__global__ void zero_kernel(float* __restrict__ p, int n) {
  int i = blockIdx.x * blockDim.x + threadIdx.x;
  if (i < n) p[i] = 0.f;
}

// ---------------- dense layer (tiny) ---------------------------------
__global__ void linear_kernel(const float* __restrict__ in, const float* __restrict__ W,
                              const float* __restrict__ bias, float* __restrict__ out,
                              int Kin, int Nout, int relu) {
  const int t = blockIdx.x * blockDim.x + threadIdx.x;
  if (t >= BB * Nout) return;
  const int b = t / Nout, j = t - b * Nout;
  float acc = bias[j];
  const float* ib = in + (long)b * Kin;
  for (int i = 0; i < Kin; ++i) acc += ib[i] * W[(long)i * Nout + j];
  out[t] = relu ? fmaxf(acc, 0.f) : acc;
}

// ---------------- final linear + log_softmax -------------------------
__global__ void head_kernel(const float* __restrict__ z2, const float* __restrict__ W,
                            const float* __restrict__ bias, float* __restrict__ out) {
  const int b = blockIdx.x * blockDim.x + threadIdx.x;
  if (b >= BB) return;
  const float* zb = z2 + (long)b * (HH / 2);
  float logits[CC];
  float mx = -3.4e38f;
  for (int c = 0; c < CC; ++c) {
    float a = bias[c];
    for (int i = 0; i < HH / 2; ++i) a += zb[i] * W[i * CC + c];
    logits[c] = a; mx = fmaxf(mx, a);
  }
  float lse = 0.f;
  for (int c = 0; c < CC; ++c) lse += expf(logits[c] - mx);
  lse = logf(lse) + mx;
  for (int c = 0; c < CC; ++c) out[b * CC + c] = logits[c] - lse;
}

// =====================================================================
extern "C" void kernel_launch(void* const* d_in, const int* in_sizes, int n_in,
                              void* d_out, int out_size, void* d_ws, size_t ws_size,
                              hipStream_t stream) {
  const float* x   = (const float*)d_in[0];
  const int*   adj = (const int*)  d_in[1];
  const float* W1  = (const float*)d_in[2];  const float* b1  = (const float*)d_in[3];
  const float* W2  = (const float*)d_in[4];  const float* b2  = (const float*)d_in[5];
  const float* W3  = (const float*)d_in[6];  const float* b3  = (const float*)d_in[7];
  const float* Wp1 = (const float*)d_in[8];  const float* bp1 = (const float*)d_in[9];
  const float* Wp2 = (const float*)d_in[10]; const float* bp2 = (const float*)d_in[11];
  const float* Wp3 = (const float*)d_in[12]; const float* bp3 = (const float*)d_in[13];
  const float* l1w = (const float*)d_in[14]; const float* l1b = (const float*)d_in[15];
  const float* l2w = (const float*)d_in[16]; const float* l2b = (const float*)d_in[17];
  const float* l3w = (const float*)d_in[18]; const float* l3b = (const float*)d_in[19];
  float* out = (float*)d_out;

  // ---- workspace bump allocator (256B aligned) ----
  char* wsc = (char*)d_ws;
  size_t off = 0;
  auto alloc = [&](size_t bytes) { size_t o = off; off += (bytes + 255) & ~(size_t)255; return o; };
  float* tbuf = (float*)(wsc + alloc((size_t)BB * NN * HH * 4));   // 64 MiB
  float* hout = (float*)(wsc + alloc((size_t)BB * NN * HH * 4));   // 64 MiB
  float* xbuf = (float*)(wsc + alloc((size_t)BB * K1 * HH * 4));   // 32 MiB
  float* a1   = (float*)(wsc + alloc((size_t)BB * K1 * K1 * 4));   // 64 MiB
  float* a2   = (float*)(wsc + alloc((size_t)BB * K2 * K2 * 4));   // 16 MiB
  float* dinv = (float*)(wsc + alloc((size_t)BB * NN * 4));
  float* svb  = (float*)(wsc + alloc((size_t)BB * NN * 4));
  float* sb   = (float*)(wsc + alloc((size_t)BB * NN * 4));
  int*   idxb = (int*)  (wsc + alloc((size_t)BB * K1 * 4));
  float* valb = (float*)(wsc + alloc((size_t)BB * K1 * 4));
  float* obuf = (float*)(wsc + alloc((size_t)BB * 2 * HH * 4));
  float* z1   = (float*)(wsc + alloc((size_t)BB * HH * 4));
  float* z2   = (float*)(wsc + alloc((size_t)BB * (HH / 2) * 4));

  zero_kernel<<<(BB * 2 * HH + 255) / 256, 256, 0, stream>>>(obuf, BB * 2 * HH);

  // =========================== Layer 1 (n=512, int adjacency) ========
  {
    const int n = NN, k = K1;
    degree_kernel<true><<<(BB * n) / 4, 128, 0, stream>>>(adj, dinv, n);
    // T = dinv (.) (X @ W1)
    wmma_gemm_kernel<false, EPI_SCALE_ROW, false>
        <<<dim3(HH / 16, n / 64, BB), 128, 0, stream>>>(
        x, W1, tbuf, n, FF, HH, (long)n * FF, 0, (long)n * HH, dinv, nullptr, nullptr);
    // H = leaky( dinv (.) (A@T + T) + b1 )
    wmma_gemm_kernel<true, EPI_GCN, true>
        <<<dim3(HH / 16, n / 64, BB), 128, 0, stream>>>(
        adj, tbuf, hout, n, n, HH, (long)n * n, (long)n * HH, (long)n * HH, dinv, tbuf, b1);
    matvec_kernel<<<(BB * n) / 4, 128, 0, stream>>>(hout, Wp1, svb);
    score_prop_kernel<true><<<(BB * n) / 4, 128, 0, stream>>>(adj, dinv, svb, bp1, sb, n);
    topk_kernel<<<BB, 256, 0, stream>>>(sb, n, k, idxb, valb);
    gather_x_kernel<<<BB * k, HH, 0, stream>>>(hout, idxb, valb, xbuf, n, k);
    gather_a_kernel<true><<<dim3(k, BB), 256, 0, stream>>>(adj, idxb, a1, n, k);
    readout_kernel<<<BB, HH, 0, stream>>>(xbuf, obuf, k);
  }

  // =========================== Layer 2 (n=256, float adjacency) ======
  {
    const int n = K1, k = K2;
    degree_kernel<false><<<(BB * n) / 4, 128, 0, stream>>>(a1, dinv, n);
    wmma_gemm_kernel<false, EPI_SCALE_ROW, false>
        <<<dim3(HH / 16, n / 64, BB), 128, 0, stream>>>(
        xbuf, W2, tbuf, n, HH, HH, (long)n * HH, 0, (long)n * HH, dinv, nullptr, nullptr);
    wmma_gemm_kernel<false, EPI_GCN, true>
        <<<dim3(HH / 16, n / 64, BB), 128, 0, stream>>>(
        a1, tbuf, hout, n, n, HH, (long)n * n, (long)n * HH, (long)n * HH, dinv, tbuf, b2);
    matvec_kernel<<<(BB * n) / 4, 128, 0, stream>>>(hout, Wp2, svb);
    score_prop_kernel<false><<<(BB * n) / 4, 128, 0, stream>>>(a1, dinv, svb, bp2, sb, n);
    topk_kernel<<<BB, 256, 0, stream>>>(sb, n, k, idxb, valb);
    gather_x_kernel<<<BB * k, HH, 0, stream>>>(hout, idxb, valb, xbuf, n, k);
    gather_a_kernel<false><<<dim3(k, BB), 128, 0, stream>>>(a1, idxb, a2, n, k);
    readout_kernel<<<BB, HH, 0, stream>>>(xbuf, obuf, k);
  }

  // =========================== Layer 3 (n=128, float adjacency) ======
  {
    const int n = K2, k = K3;
    degree_kernel<false><<<(BB * n) / 4, 128, 0, stream>>>(a2, dinv, n);
    wmma_gemm_kernel<false, EPI_SCALE_ROW, false>
        <<<dim3(HH / 16, n / 64, BB), 128, 0, stream>>>(
        xbuf, W3, tbuf, n, HH, HH, (long)n * HH, 0, (long)n * HH, dinv, nullptr, nullptr);
    wmma_gemm_kernel<false, EPI_GCN, true>
        <<<dim3(HH / 16, n / 64, BB), 128, 0, stream>>>(
        a2, tbuf, hout, n, n, HH, (long)n * n, (long)n * HH, (long)n * HH, dinv, tbuf, b3);
    matvec_kernel<<<(BB * n) / 4, 128, 0, stream>>>(hout, Wp3, svb);
    score_prop_kernel<false><<<(BB * n) / 4, 128, 0, stream>>>(a2, dinv, svb, bp3, sb, n);
    topk_kernel<<<BB, 256, 0, stream>>>(sb, n, k, idxb, valb);
    gather_x_kernel<<<BB * k, HH, 0, stream>>>(hout, idxb, valb, xbuf, n, k);
    readout_kernel<<<BB, HH, 0, stream>>>(xbuf, obuf, k);
  }

  // =========================== MLP head ==============================
  linear_kernel<<<(BB * HH + 255) / 256, 256, 0, stream>>>(obuf, l1w, l1b, z1, 2 * HH, HH, 1);
  linear_kernel<<<(BB * (HH / 2) + 255) / 256, 256, 0, stream>>>(z1, l2w, l2b, z2, HH, HH / 2, 1);
  head_kernel<<<1, 256, 0, stream>>>(z2, l3w, l3b, out);
}